// yolo_loss_74955769249863
// MI455X (gfx1250) — compile-verified
//
#include <hip/hip_runtime.h>
#include <math.h>

// ---------------- problem constants (from reference) ----------------
#define NC        20
#define PRED_CH   30              // NC + 2*5
#define TGT_CH    25              // NC + 5
#define NCELLS    (16384 * 49)    // 802,816 cells, contiguous AoS in memory
#define TPB       256             // 8 waves of 32 on CDNA5
#define NBLOCKS   (NCELLS / TPB)  // 3136, exact
#define LAMBDA_COORD 5.0f
#define LAMBDA_NOOBJ 0.5f
#define EPSF      1e-6f

#define PRED_F4   (TPB * PRED_CH / 4)  // 1920 float4 per block tile (30,720 B)
#define TGT_F4    (TPB * TGT_CH  / 4)  // 1600 float4 per block tile (25,600 B)

typedef __attribute__((ext_vector_type(2))) float v2f;
typedef __attribute__((ext_vector_type(4))) float f4;
typedef __attribute__((ext_vector_type(8))) float v8f;
typedef __attribute__((ext_vector_type(4))) unsigned int v4u;
typedef __attribute__((ext_vector_type(8))) int v8i;
typedef __attribute__((ext_vector_type(4))) int v4i;

#if __has_builtin(__builtin_amdgcn_tensor_load_to_lds) && __has_builtin(__builtin_amdgcn_s_wait_tensorcnt)
#define USE_TDM 1
#endif

#ifdef USE_TDM
// DMA a contiguous 1-D span (n8 x 8-byte elements) from global into LDS via the
// Tensor Data Mover. D# layout per CDNA5 ISA ch.8 (CDNA5 ISA):
//   group0: [1:0]=count=1, [63:32]=lds_addr, [120:64]=global_addr, [127:126]=type=2
//   group1: [17:16]=data_size=3(8B), [79:48]=tensor_dim0, [95:80]=tensor_dim1=1,
//           [127:112]=tile_dim0, [143:128]=tile_dim1=1, [207:160]=tensor_dim0_stride
// 6-arg clang-23 builtin: (v4u g0, v8i g1, v4i g2, v4i g3, v8i extra, i32 cpol).
// Issued by one wave only; tracked by TENSORcnt.
__device__ __forceinline__ void tdm_load_1d(unsigned lds_off, const void* gptr, unsigned n8) {
  unsigned long long ga = (unsigned long long)(uintptr_t)gptr;
  v4u g0;
  g0.x = 1u;                                            // count=1 (valid user D#)
  g0.y = lds_off;                                       // LDS byte address
  g0.z = (unsigned)(ga & 0xFFFFFFFFu);                  // global_addr[31:0]
  g0.w = (unsigned)((ga >> 32) & 0x01FFFFFFu) | (2u << 30); // global_addr[56:32] | type=2
  v8i g1;
  g1[0] = (int)(3u << 16);                              // data_size=3 (8B); no pad/iter/mcast
  g1[1] = (int)((n8 & 0xFFFFu) << 16);                  // tensor_dim0[15:0]
  g1[2] = (int)((n8 >> 16) | (1u << 16));               // tensor_dim0[31:16], tensor_dim1=1
  g1[3] = (int)((n8 & 0xFFFFu) << 16);                  // tile_dim0 = n8
  g1[4] = 1;                                            // tile_dim1=1, tile_dim2=0
  g1[5] = (int)n8;                                      // tensor_dim0_stride[31:0]
  g1[6] = 0;                                            // stride hi, tensor_dim1_stride lo
  g1[7] = 0;
  v4i z4 = {0, 0, 0, 0};                                // groups 2/3 unused (<=2D)
  v8i z8 = {0, 0, 0, 0, 0, 0, 0, 0};                    // extra group (unused)
  __builtin_amdgcn_tensor_load_to_lds(g0, g1, z4, z4, z8, 0);
}
#endif

__device__ __forceinline__ float iou_mid(float bx, float by, float bw, float bh,
                                         float tx, float ty, float tw, float th) {
  float b_x1 = bx - bw * 0.5f, b_x2 = bx + bw * 0.5f;
  float b_y1 = by - bh * 0.5f, b_y2 = by + bh * 0.5f;
  float t_x1 = tx - tw * 0.5f, t_x2 = tx + tw * 0.5f;
  float t_y1 = ty - th * 0.5f, t_y2 = ty + th * 0.5f;
  float iw = fmaxf(fminf(b_x2, t_x2) - fmaxf(b_x1, t_x1), 0.0f);
  float ih = fmaxf(fminf(b_y2, t_y2) - fmaxf(b_y1, t_y1), 0.0f);
  float inter = iw * ih;
  float a1 = fabsf((b_x2 - b_x1) * (b_y2 - b_y1));
  float a2 = fabsf((t_x2 - t_x1) * (t_y2 - t_y1));
  return inter / (a1 + a2 - inter + EPSF);
}

__global__ __launch_bounds__(TPB) void yolo_loss_main(const float* __restrict__ pred,
                                                      const float* __restrict__ tgt,
                                                      float* __restrict__ partial) {
  __shared__ f4 sP4[PRED_F4];       // 30,720 B
  __shared__ f4 sT4[TGT_F4];        // 25,600 B
  __shared__ float wsum[TPB / 32];

  const int tid = threadIdx.x;

#ifdef USE_TDM
  // ---- TDM DMA staging: one tensor_load_to_lds per tile, issued by wave 0 ----
  if (tid == 0) {
    const float* gp = pred + (size_t)blockIdx.x * (PRED_F4 * 4);
    const float* gt = tgt  + (size_t)blockIdx.x * (TGT_F4 * 4);
    tdm_load_1d((unsigned)(uintptr_t)(void*)sP4, gp, PRED_F4 * 2);  // 3840 x 8B
    tdm_load_1d((unsigned)(uintptr_t)(void*)sT4, gt, TGT_F4 * 2);   // 3200 x 8B
    __builtin_amdgcn_s_wait_tensorcnt(0);
  }
  __syncthreads();
#else
  // ---- fallback: coalesced nontemporal b128 staging ----
  {
    const f4* gp = reinterpret_cast<const f4*>(pred) + (size_t)blockIdx.x * PRED_F4;
    #pragma unroll
    for (int i = tid; i < PRED_F4; i += TPB) sP4[i] = __builtin_nontemporal_load(gp + i);
    const f4* gt = reinterpret_cast<const f4*>(tgt) + (size_t)blockIdx.x * TGT_F4;
    #pragma unroll
    for (int i = tid; i < TGT_F4; i += TPB) sT4[i] = __builtin_nontemporal_load(gt + i);
  }
  __syncthreads();
#endif

  // ---- per-thread cell from LDS (stride 30/25 dwords: conflict-light) ----
  const float* P = reinterpret_cast<const float*>(sP4) + tid * PRED_CH;
  const float* T = reinterpret_cast<const float*>(sT4) + tid * TGT_CH;

  const float obj = T[20];                 // == p_t
  const float tbx = T[21], tby = T[22], tbw = T[23], tbh = T[24];

  const float c0 = P[20], x0 = P[21], y0 = P[22], w0 = P[23], h0 = P[24];
  const float c1 = P[25], x1 = P[26], y1 = P[27], w1 = P[28], h1 = P[29];

  const float iou0 = iou_mid(x0, y0, w0, h0, tbx, tby, tbw, tbh);
  const float iou1 = iou_mid(x1, y1, w1, h1, tbx, tby, tbw, tbh);
  const bool pick1 = iou1 > iou0;          // argmax, first-max tie -> box0

  const float bc = pick1 ? c1 : c0;
  const float bx = pick1 ? x1 : x0;
  const float by = pick1 ? y1 : y0;
  const float bw = pick1 ? w1 : w0;
  const float bh = pick1 ? h1 : h0;

  const float sgw = (float)((bw > 0.0f) - (bw < 0.0f));
  const float sgh = (float)((bh > 0.0f) - (bh < 0.0f));
  const float sw = sgw * sqrtf(fabsf(bw) + EPSF);
  const float sh = sgh * sqrtf(fabsf(bh) + EPSF);

  const float d0 = obj * bx - obj * tbx;
  const float d1 = obj * by - obj * tby;
  const float d2 = obj * sw - sqrtf(obj * tbw);
  const float d3 = obj * sh - sqrtf(obj * tbh);
  const float box_loss = d0 * d0 + d1 * d1 + d2 * d2 + d3 * d3;

  const float noobj = 1.0f - obj;
  const float e0 = noobj * (c0 - obj), e1 = noobj * (c1 - obj);
  const float no_obj_loss = e0 * e0 + e1 * e1;

  const float eo = obj * (bc - obj);
  const float obj_loss = eo * eo;

  float class_loss = 0.0f;
  #pragma unroll
  for (int k = 0; k < NC; ++k) {
    const float dc = obj * (P[k] - T[k]);
    class_loss += dc * dc;
  }

  float acc = LAMBDA_COORD * box_loss + obj_loss + LAMBDA_NOOBJ * no_obj_loss + class_loss;

  // ---- wave reduction via V_WMMA_F32_16X16X4_F32 against an all-ones B ----
  // A.x=acc, A.y=0 => D[m][n] = acc_m + acc_{m+16} (exact f32); sum 8 D comps +
  // one xor-16 shuffle = full 32-lane sum.
  float s;
#if defined(__has_builtin) && __has_builtin(__builtin_amdgcn_wmma_f32_16x16x4_f32)
  {
    v2f a;  a.x = acc;  a.y = 0.0f;
    v2f b1; b1.x = 1.0f; b1.y = 1.0f;
    v8f cz = {};
    v8f dm = __builtin_amdgcn_wmma_f32_16x16x4_f32(
        /*neg_a=*/false, a, /*neg_b=*/false, b1,
        /*c_mod=*/(short)0, cz, /*reuse_a=*/false, /*reuse_b=*/false);
    s = ((dm[0] + dm[1]) + (dm[2] + dm[3])) + ((dm[4] + dm[5]) + (dm[6] + dm[7]));
    s += __shfl_xor(s, 16, 32);
  }
#else
  s = acc;
  #pragma unroll
  for (int off = 16; off > 0; off >>= 1) s += __shfl_xor(s, off, 32);
#endif

  // ---- block combine, one float of output traffic per block ----
  const int lane = tid & 31;
  const int wave = tid >> 5;
  if (lane == 0) wsum[wave] = s;
  __syncthreads();
  if (tid == 0) {
    float b = 0.0f;
    #pragma unroll
    for (int w = 0; w < TPB / 32; ++w) b += wsum[w];
    partial[blockIdx.x] = b;
  }
}

// Deterministic fixed-order second pass: 3136 partials -> scalar.
__global__ __launch_bounds__(TPB) void yolo_loss_reduce(const float* __restrict__ partial,
                                                        float* __restrict__ out, int n) {
  __shared__ float red[TPB];
  float s = 0.0f;
  for (int i = threadIdx.x; i < n; i += TPB) s += partial[i];
  red[threadIdx.x] = s;
  __syncthreads();
  for (int off = TPB / 2; off > 0; off >>= 1) {
    if ((int)threadIdx.x < off) red[threadIdx.x] += red[threadIdx.x + off];
    __syncthreads();
  }
  if (threadIdx.x == 0) out[0] = red[0];
}

extern "C" void kernel_launch(void* const* d_in, const int* in_sizes, int n_in,
                              void* d_out, int out_size, void* d_ws, size_t ws_size,
                              hipStream_t stream) {
  const float* pred = (const float*)d_in[0];   // (16384, 7, 7, 30) f32, contiguous
  const float* tgt  = (const float*)d_in[1];   // (16384, 7, 7, 25) f32, contiguous
  float* partial = (float*)d_ws;               // NBLOCKS floats of scratch

  yolo_loss_main<<<NBLOCKS, TPB, 0, stream>>>(pred, tgt, partial);
  yolo_loss_reduce<<<1, TPB, 0, stream>>>(partial, (float*)d_out, NBLOCKS);
}